// GATv2Layer_30468497998544
// MI455X (gfx1250) — compile-verified
//
#include <hip/hip_runtime.h>
#include <math.h>

#define HEADS     4
#define F_OUT     64
#define N_NODES   4096
#define F_IN      256
#define F_TOT     256      // F_OUT * HEADS
#define NEG_SLOPE 0.2f

typedef __attribute__((ext_vector_type(2))) float v2f;
typedef __attribute__((ext_vector_type(8))) float v8f;

static __device__ __forceinline__ v8f wmma_f32(v2f a, v2f b, v8f c) {
  // D = A(16x4 f32) * B(4x16 f32) + C(16x16 f32), full f32 precision
  return __builtin_amdgcn_wmma_f32_16x16x4_f32(
      /*neg_a=*/false, a, /*neg_b=*/false, b,
      /*c_mod=*/(short)0, c, /*reuse_a=*/false, /*reuse_b=*/false);
}

// ---- async global->LDS staging (CDNA5 ASYNCcnt path) with safe fallback ----
#if defined(__gfx1250__) && \
    __has_builtin(__builtin_amdgcn_global_load_async_to_lds_b128) && \
    __has_builtin(__builtin_amdgcn_s_wait_asynccnt)
#define USE_ASYNC_LDS 1
#else
#define USE_ASYNC_LDS 0
#endif

#if USE_ASYNC_LDS
typedef int v4i_16 __attribute__((vector_size(16)));
typedef __attribute__((address_space(1))) v4i_16 g_v4i;   // global
typedef __attribute__((address_space(3))) v4i_16 l_v4i;   // LDS
static __device__ __forceinline__ void copy16(const void* gsrc, void* ldst) {
  __builtin_amdgcn_global_load_async_to_lds_b128((g_v4i*)gsrc, (l_v4i*)ldst,
                                                 /*offset=*/0, /*cpol=*/0);
}
static __device__ __forceinline__ void async_wait0() {
  __builtin_amdgcn_s_wait_asynccnt(0);
}
#else
static __device__ __forceinline__ void copy16(const void* gsrc, void* ldst) {
  *(int4*)ldst = *(const int4*)gsrc;
}
static __device__ __forceinline__ void async_wait0() {}
#endif

// ---------------------------------------------------------------------------
// Kernel 1: h = x @ W   (4096x256 · 256x256, f32 WMMA 16x16x4)
// ---------------------------------------------------------------------------
__global__ __launch_bounds__(128) void k_gemm_h(const float* __restrict__ x,
                                                const float* __restrict__ W,
                                                float* __restrict__ h) {
  const int lane = threadIdx.x & 31;
  const int wave = threadIdx.x >> 5;
  const int tile = blockIdx.x * 4 + wave;   // 4096 tiles total
  const int i0 = (tile >> 4) * 16;
  const int n0 = (tile & 15) * 16;
  const int row = lane & 15;
  const int kh  = lane >> 4;

  v8f acc = {0.f, 0.f, 0.f, 0.f, 0.f, 0.f, 0.f, 0.f};
  const float* xrow = x + (size_t)(i0 + row) * F_IN;

#pragma unroll 4
  for (int k0 = 0; k0 < F_IN; k0 += 4) {
    const int ka = k0 + kh * 2;
    v2f a = *(const v2f*)(xrow + ka);       // A: (M=row, K=ka, ka+1)
    v2f b;                                  // B: (K=ka(+1), N=row)
    b.x = W[(size_t)(ka + 0) * F_TOT + n0 + row];
    b.y = W[(size_t)(ka + 1) * F_TOT + n0 + row];
    acc = wmma_f32(a, b, acc);
  }

#pragma unroll
  for (int r = 0; r < 8; ++r)               // C/D: M = r + 8*kh, N = row
    h[(size_t)(i0 + r + 8 * kh) * F_TOT + n0 + row] = acc[r];
}

// ---------------------------------------------------------------------------
// Kernel 2: el[n,hd] = h[n,hd,:]·a1   er[n,hd] = h[n,hd,:]·a2
// ---------------------------------------------------------------------------
__global__ __launch_bounds__(256) void k_el_er(const float* __restrict__ h,
                                               const float* __restrict__ a,
                                               float* __restrict__ el,
                                               float* __restrict__ er) {
  const int idx = blockIdx.x * 256 + threadIdx.x;   // 16384 threads
  const int n  = idx >> 2;
  const int hd = idx & 3;
  const float* hp = h + (size_t)n * F_TOT + hd * F_OUT;
  float s1 = 0.f, s2 = 0.f;
#pragma unroll 8
  for (int f = 0; f < F_OUT; ++f) {
    float v = hp[f];
    s1 = fmaf(v, a[f], s1);
    s2 = fmaf(v, a[F_OUT + f], s2);
  }
  el[idx] = s1;
  er[idx] = s2;
}

// ---------------------------------------------------------------------------
// Kernel 3: fused masked-softmax + aggregation + head-mean.
//   128 blocks x 512 threads. Block = 32 output rows.
//   16 waves = 4 heads x 4 j-quarters (1024 j each).
//   Per wave: two 16-row A tiles (p weights), 8 f32 WMMA accumulators,
//   unnormalized softmax (exp bounded, no max pass), per-lane exp-sums.
//   adj + er tiles double-buffered in LDS via async global->LDS loads.
//   Epilogue: ds_add_f32 combine across j-quarters, normalize, head-mean.
// ---------------------------------------------------------------------------
#define QTR_J   1024               // j range per quarter
#define CHUNK_J 64                 // j per staged chunk
#define ROWS    32                 // i rows per block

static __device__ __forceinline__ void stage_chunk(const int* __restrict__ adj,
                                                   const float* __restrict__ er,
                                                   int i0, int k, int t,
                                                   int* __restrict__ adjt,
                                                   float* __restrict__ erb) {
  // adj: 4 quarters x 32 rows x 64 ints = 8192 ints = 2048 x 16B
#pragma unroll
  for (int u = 0; u < 4; ++u) {
    const int idx = u * 512 + t;
    const int qtr = idx >> 9;
    const int rem = idx & 511;
    const int r   = rem >> 4;
    const int c4  = rem & 15;
    const int* src = adj + (size_t)(i0 + r) * N_NODES + qtr * QTR_J + k * CHUNK_J + c4 * 4;
    int* dst = adjt + qtr * (ROWS * CHUNK_J) + r * CHUNK_J + c4 * 4;
    copy16(src, dst);
  }
  // er: 4 quarters x 64 j x 4 heads floats = 1024 floats = 256 x 16B
  if (t < 256) {
    const int qtr = t >> 6, j = t & 63;
    const float* src = er + (size_t)(qtr * QTR_J + k * CHUNK_J + j) * HEADS;
    float* dst = erb + qtr * (CHUNK_J * HEADS) + j * HEADS;
    copy16(src, dst);
  }
}

__global__ __launch_bounds__(512) void k_aggregate(const float* __restrict__ h,
                                                   const float* __restrict__ el,
                                                   const float* __restrict__ er,
                                                   const int* __restrict__ adj,
                                                   float* __restrict__ out) {
  __shared__ __align__(16) int   adjt[2][HEADS][ROWS * CHUNK_J];   // 64 KB
  __shared__ __align__(16) float erb [2][HEADS][CHUNK_J * HEADS];  //  8 KB
  __shared__ float outb[HEADS][ROWS * F_OUT];                      // 32 KB
  __shared__ float ssb [HEADS][ROWS];                              // 0.5 KB

  const int t    = threadIdx.x;
  const int lane = t & 31;
  const int w    = t >> 5;
  const int hd   = w & 3;            // head
  const int qt   = w >> 2;           // j-quarter
  const int row  = lane & 15;
  const int kh   = lane >> 4;
  const int kh2  = kh * 2;
  const int i0   = blockIdx.x * ROWS;

  const float el0 = el[(size_t)(i0 + row) * HEADS + hd];
  const float el1 = el[(size_t)(i0 + 16 + row) * HEADS + hd];

  // zero epilogue buffers (visibility ordered by in-loop barriers)
#pragma unroll
  for (int u = 0; u < 4; ++u) outb[0][u * 512 + t] = 0.f;   // 4*512 = 2048 floats, outb[0] only
  if (t < HEADS * ROWS) ssb[0][t] = 0.f;

  v8f accs[8];
#pragma unroll
  for (int q = 0; q < 8; ++q) {
    v8f z = {0.f, 0.f, 0.f, 0.f, 0.f, 0.f, 0.f, 0.f};
    accs[q] = z;
  }
  float ssum0 = 0.f, ssum1 = 0.f;

  stage_chunk(adj, er, i0, 0, t, &adjt[0][0][0], &erb[0][0][0]);

  const int NCHUNK = QTR_J / CHUNK_J;   // 16
  for (int k = 0; k < NCHUNK; ++k) {
    const int cur = k & 1;
    async_wait0();                      // my async loads landed in LDS
    __syncthreads();                    // everyone's landed, prev reads done
    if (k + 1 < NCHUNK)
      stage_chunk(adj, er, i0, k + 1, t, &adjt[1 - cur][0][0], &erb[1 - cur][0][0]);

    const int*   at = &adjt[cur][qt][0];
    const float* eb = &erb[cur][qt][0];
    const int jbase = qt * QTR_J + k * CHUNK_J;

#pragma unroll 2
    for (int j0 = 0; j0 < CHUNK_J; j0 += 4) {
      const int ja = j0 + kh2;          // this lane's local K pair
      const float er0 = eb[ja * HEADS + hd];
      const float er1 = eb[ja * HEADS + HEADS + hd];
      const int a0 = at[row * CHUNK_J + ja];
      const int a1 = at[row * CHUNK_J + ja + 1];
      const int b0 = at[(row + 16) * CHUNK_J + ja];
      const int b1 = at[(row + 16) * CHUNK_J + ja + 1];

      float e;
      e = el0 + er0; e = e > 0.f ? e : NEG_SLOPE * e;
      const float p00 = a0 ? __expf(e) : 0.f;
      e = el0 + er1; e = e > 0.f ? e : NEG_SLOPE * e;
      const float p01 = a1 ? __expf(e) : 0.f;
      e = el1 + er0; e = e > 0.f ? e : NEG_SLOPE * e;
      const float p10 = b0 ? __expf(e) : 0.f;
      e = el1 + er1; e = e > 0.f ? e : NEG_SLOPE * e;
      const float p11 = b1 ? __expf(e) : 0.f;

      ssum0 += p00 + p01;
      ssum1 += p10 + p11;
      v2f A0; A0.x = p00; A0.y = p01;
      v2f A1; A1.x = p10; A1.y = p11;

      const float* hb0 = h + (size_t)(jbase + ja) * F_TOT + hd * F_OUT + row;
      const float* hb1 = hb0 + F_TOT;
#pragma unroll
      for (int ft = 0; ft < 4; ++ft) {
        v2f B;                          // B reused by both A tiles (registers)
        B.x = hb0[ft * 16];
        B.y = hb1[ft * 16];
        accs[ft]     = wmma_f32(A0, B, accs[ft]);
        accs[4 + ft] = wmma_f32(A1, B, accs[4 + ft]);
      }
    }
  }

  // ---- combine j-quarters (LDS float atomics), normalize, head-mean ----
#pragma unroll
  for (int ft = 0; ft < 4; ++ft) {
#pragma unroll
    for (int r = 0; r < 8; ++r) {
      const int M = r + 8 * kh;
      atomicAdd(&outb[hd][M * F_OUT + ft * 16 + row], accs[ft][r]);
      atomicAdd(&outb[hd][(16 + M) * F_OUT + ft * 16 + row], accs[4 + ft][r]);
    }
  }
  atomicAdd(&ssb[hd][row], ssum0);
  atomicAdd(&ssb[hd][16 + row], ssum1);
  __syncthreads();

#pragma unroll
  for (int u = 0; u < 4; ++u) {
    const int idx = u * 512 + t;        // 32*64 = 2048 outputs
    const int r = idx >> 6, f = idx & 63;
    float v = 0.f;
#pragma unroll
    for (int hh = 0; hh < HEADS; ++hh)
      v += outb[hh][r * F_OUT + f] / ssb[hh][r];
    out[(size_t)(i0 + r) * F_OUT + f] = 0.25f * v;
  }
}

// ---------------------------------------------------------------------------
extern "C" void kernel_launch(void* const* d_in, const int* in_sizes, int n_in,
                              void* d_out, int out_size, void* d_ws, size_t ws_size,
                              hipStream_t stream) {
  const float* x   = (const float*)d_in[0];
  const int*   adj = (const int*)d_in[1];
  const float* W   = (const float*)d_in[2];
  const float* a   = (const float*)d_in[3];
  float* out = (float*)d_out;

  // workspace layout: h (4 MB) | el (64 KB) | er (64 KB)
  float* h  = (float*)d_ws;
  float* el = h + (size_t)N_NODES * F_TOT;
  float* er = el + (size_t)N_NODES * HEADS;

  k_gemm_h   <<<1024, 128, 0, stream>>>(x, W, h);
  k_el_er    <<<64,   256, 0, stream>>>(h, a, el, er);
  k_aggregate<<<128,  512, 0, stream>>>(h, el, er, adj, out);
}